// _TinyModel_52785148067862
// MI455X (gfx1250) — compile-verified
//
#include <hip/hip_runtime.h>

// Sizes from the reference
#define DIM     4
#define VOCAB   16
#define BSZ     256
#define TLEN    8192
#define CHUNK   64                 // chunk length for the parallel scan (2^6)
#define NCHUNK  (TLEN / CHUNK)     // 128 chunks per row
#define NGROUP  (BSZ / 16)         // 16 groups of 16 batch rows (one WMMA A tile)

typedef __attribute__((ext_vector_type(2))) float v2f;
typedef __attribute__((ext_vector_type(8))) float v8f;

// ---------------------------------------------------------------------------
// Pass 1: per-(row, chunk) local scan with zero initial state.
// local_out[(row*NCHUNK + chunk)*4 + i] = end-state of chunk (zero init).
// 32768 independent threads; each does 64 sequential 4x4 matvecs in registers.
// ---------------------------------------------------------------------------
__global__ __launch_bounds__(256)
void scan_local(const int* __restrict__ ids,
                const float* __restrict__ embed,
                const float* __restrict__ W,
                float* __restrict__ local_out) {
    __shared__ __align__(16) float s_emb[VOCAB * DIM];
    const int tid = threadIdx.x;
    if (tid < VOCAB * DIM) s_emb[tid] = embed[tid];
    __syncthreads();

    float w[16];
#pragma unroll
    for (int i = 0; i < 16; ++i) w[i] = W[i];

    const int task  = blockIdx.x * blockDim.x + tid;   // row*NCHUNK + chunk
    const int row   = task >> 7;                       // / NCHUNK
    const int chunk = task & (NCHUNK - 1);
    const int* idp  = ids + row * TLEN + chunk * CHUNK;

    float s0 = 0.f, s1 = 0.f, s2 = 0.f, s3 = 0.f;
    for (int t = 0; t < CHUNK; ++t) {
        const int id = idp[t];
        const float4 h = *reinterpret_cast<const float4*>(&s_emb[id * DIM]);
        // depth-3 trees instead of depth-4 chains
        float n0 = fmaf(w[ 1], s1, w[ 0] * s0) + fmaf(w[ 3], s3, fmaf(w[ 2], s2, h.x));
        float n1 = fmaf(w[ 5], s1, w[ 4] * s0) + fmaf(w[ 7], s3, fmaf(w[ 6], s2, h.y));
        float n2 = fmaf(w[ 9], s1, w[ 8] * s0) + fmaf(w[11], s3, fmaf(w[10], s2, h.z));
        float n3 = fmaf(w[13], s1, w[12] * s0) + fmaf(w[15], s3, fmaf(w[14], s2, h.w));
        s0 = n0; s1 = n1; s2 = n2; s3 = n3;
    }
    *reinterpret_cast<float4*>(local_out + (size_t)task * 4) = make_float4(s0, s1, s2, s3);
}

// ---------------------------------------------------------------------------
// Pass 2: per-row sequential combine across chunks.
//   in_{c+1} = W^CHUNK * in_c + local_c ;  carry_out[row][c] = in_c
// W^64 computed per-thread with 6 register-level 4x4 squarings.
// ---------------------------------------------------------------------------
__global__ __launch_bounds__(256)
void scan_combine(const float* __restrict__ W,
                  const float* __restrict__ local_in,
                  float* __restrict__ carry_out) {
    const int row = threadIdx.x;            // one block of 256 threads

    float P[16];
#pragma unroll
    for (int i = 0; i < 16; ++i) P[i] = W[i];
#pragma unroll
    for (int sq = 0; sq < 6; ++sq) {        // P = W^(2^6) = W^64
        float Q[16];
#pragma unroll
        for (int i = 0; i < 4; ++i)
#pragma unroll
            for (int j = 0; j < 4; ++j) {
                float acc = 0.f;
#pragma unroll
                for (int k = 0; k < 4; ++k) acc = fmaf(P[i*4+k], P[k*4+j], acc);
                Q[i*4+j] = acc;
            }
#pragma unroll
        for (int i = 0; i < 16; ++i) P[i] = Q[i];
    }

    float c0 = 0.f, c1 = 0.f, c2 = 0.f, c3 = 0.f;
    for (int c = 0; c < NCHUNK; ++c) {
        const size_t idx = ((size_t)row * NCHUNK + c) * 4;
        *reinterpret_cast<float4*>(carry_out + idx) = make_float4(c0, c1, c2, c3);
        const float4 l = *reinterpret_cast<const float4*>(local_in + idx);
        float n0 = fmaf(P[ 1], c1, P[ 0] * c0) + fmaf(P[ 3], c3, fmaf(P[ 2], c2, l.x));
        float n1 = fmaf(P[ 5], c1, P[ 4] * c0) + fmaf(P[ 7], c3, fmaf(P[ 6], c2, l.y));
        float n2 = fmaf(P[ 9], c1, P[ 8] * c0) + fmaf(P[11], c3, fmaf(P[10], c2, l.z));
        float n3 = fmaf(P[13], c1, P[12] * c0) + fmaf(P[15], c3, fmaf(P[14], c2, l.w));
        c0 = n0; c1 = n1; c2 = n2; c3 = n3;
    }
}

// ---------------------------------------------------------------------------
// Pass 3: fused re-scan + logits via V_WMMA_F32_16X16X4_F32.
// One wave owns (group of 16 batch rows, chunk of 64 timesteps).
// Lane l: m = l&15 (batch row in group), kp = l>>4.
// Every lane carries the FULL 4-component state of its row (lanes m and m+16
// redundantly) -> no cross-lane shuffle on the recurrence critical path.
// The WMMA A operand (f32 16x4 layout: V0 = K0/K2, V1 = K1/K3) is just a
// 2-cndmask selection of the lane's K-pair from the redundant state.
// Each step: 16 dual-issuable FMAs, then one WMMA
//   D[16x16] = A(state 16x4) x B(lm_W^T 4x16) + C(bias broadcast).
// ---------------------------------------------------------------------------
__global__ __launch_bounds__(256)
void scan_emit_wmma(const int* __restrict__ ids,
                    const float* __restrict__ embed,
                    const float* __restrict__ W,
                    const float* __restrict__ lm_W,
                    const float* __restrict__ lm_b,
                    const float* __restrict__ carry,
                    float* __restrict__ out) {
    __shared__ __align__(16) float s_emb[VOCAB * DIM];
    const int tid = threadIdx.x;
    if (tid < VOCAB * DIM) s_emb[tid] = embed[tid];
    __syncthreads();

    const int wave = tid >> 5;
    const int lane = tid & 31;
    const int task = blockIdx.x * 8 + wave;       // g*NCHUNK + chunk, 2048 total
    const int g     = task >> 7;                  // / NCHUNK
    const int chunk = task & (NCHUNK - 1);

    const int m  = lane & 15;                     // row-in-group == vocab column n
    const bool hi = (lane >> 4) != 0;             // which K pair this lane supplies
    const int row = g * 16 + m;

    float w[16];
#pragma unroll
    for (int i = 0; i < 16; ++i) w[i] = W[i];

    // B operand: B(k,n) = lm_W[n][k]; lane -> n = m, k = 2*kp, 2*kp+1
    v2f bmat;
    {
        const int kb = hi ? 2 : 0;
        bmat.x = lm_W[m * DIM + kb];
        bmat.y = lm_W[m * DIM + kb + 1];
    }

    // C operand: bias broadcast (C[M][N] = lm_b[N], N = lane&15)
    const float bias = lm_b[m];
    v8f cmat;
#pragma unroll
    for (int i = 0; i < 8; ++i) cmat[i] = bias;

    // incoming full state for this chunk (redundant across the two half-waves)
    const float4 cin = *reinterpret_cast<const float4*>(
        carry + ((size_t)row * NCHUNK + chunk) * 4);
    float s0 = cin.x, s1 = cin.y, s2 = cin.z, s3 = cin.w;

    const int* idp = ids + (size_t)row * TLEN + (size_t)chunk * CHUNK;
    const size_t tbase = (size_t)chunk * CHUNK;

    for (int t = 0; t < CHUNK; ++t) {
        const int id = idp[t];
        const float4 h = *reinterpret_cast<const float4*>(&s_emb[id * DIM]);

        // s = W*s + h  (full 4-vector per lane; depth-3 FMA trees, dual-issue)
        float n0 = fmaf(w[ 1], s1, w[ 0] * s0) + fmaf(w[ 3], s3, fmaf(w[ 2], s2, h.x));
        float n1 = fmaf(w[ 5], s1, w[ 4] * s0) + fmaf(w[ 7], s3, fmaf(w[ 6], s2, h.y));
        float n2 = fmaf(w[ 9], s1, w[ 8] * s0) + fmaf(w[11], s3, fmaf(w[10], s2, h.z));
        float n3 = fmaf(w[13], s1, w[12] * s0) + fmaf(w[15], s3, fmaf(w[14], s2, h.w));
        s0 = n0; s1 = n1; s2 = n2; s3 = n3;

        // A operand: lanes 0-15 supply (K0,K1), lanes 16-31 supply (K2,K3)
        v2f a;
        a.x = hi ? s2 : s0;
        a.y = hi ? s3 : s1;

        v8f d = __builtin_amdgcn_wmma_f32_16x16x4_f32(
            /*neg_a=*/false, a, /*neg_b=*/false, bmat,
            /*c_mod=*/(short)0, cmat, /*reuse_a=*/false, /*reuse_b=*/false);

        // D: VGPR i -> M = i (lanes 0-15) / 8+i (lanes 16-31); N = lane&15
        const size_t tg = tbase + t;
        const size_t obase = ((size_t)(g * 16 + (hi ? 8 : 0)) * TLEN + tg) * VOCAB + m;
#pragma unroll
        for (int i = 0; i < 8; ++i) {
            out[obase + (size_t)i * TLEN * VOCAB] = d[i];
        }
    }
}

// ---------------------------------------------------------------------------
extern "C" void kernel_launch(void* const* d_in, const int* in_sizes, int n_in,
                              void* d_out, int out_size, void* d_ws, size_t ws_size,
                              hipStream_t stream) {
    (void)in_sizes; (void)n_in; (void)out_size; (void)ws_size;
    const int*   ids   = (const int*)  d_in[0];   // [256, 8192]
    const float* embed = (const float*)d_in[1];   // [16, 4]
    const float* W     = (const float*)d_in[2];   // [4, 4]
    const float* lm_W  = (const float*)d_in[3];   // [16, 4]
    const float* lm_b  = (const float*)d_in[4];   // [16]
    float*       out   = (float*)d_out;           // [256, 8192, 16]

    float* ws_local = (float*)d_ws;                         // 256*128*4 floats
    float* ws_carry = ws_local + (size_t)BSZ * NCHUNK * 4;  // 256*128*4 floats

    // Pass 1: 32768 (row, chunk) tasks
    scan_local<<<(BSZ * NCHUNK) / 256, 256, 0, stream>>>(ids, embed, W, ws_local);
    // Pass 2: one thread per batch row
    scan_combine<<<1, 256, 0, stream>>>(W, ws_local, ws_carry);
    // Pass 3: 2048 wave-tasks (16 row-groups x 128 chunks), 8 waves/block
    scan_emit_wmma<<<(NGROUP * NCHUNK) / 8, 256, 0, stream>>>(
        ids, embed, W, lm_W, lm_b, ws_carry, out);
}